// BaseLoss_6382321402276
// MI455X (gfx1250) — compile-verified
//
#include <hip/hip_runtime.h>
#include <hip/hip_bf16.h>

// ---------------------------------------------------------------------------
// loss = (1/(2B)) * sum_b [ lse(start_b) - start_b[sp] + lse(end_b) - end_b[ep] ]
// B = 256, S = 32768.  Memory-bound: 64 MB of logits -> ~2.8 us at 23.3 TB/s.
// Kernel 1: 512 blocks, each computes one row's (logsumexp - x[pos]) partial.
// Kernel 2: one wave32 reduces the 512 partials with V_WMMA_F32_16X16X4_F32.
// ---------------------------------------------------------------------------

typedef __attribute__((ext_vector_type(2))) float v2f;
typedef __attribute__((ext_vector_type(8))) float v8f;

#define SEQ 32768
#define NTHREADS 256
#define NWAVES (NTHREADS / 32)     // 8 wave32 waves per block
#define CHUNKS (SEQ / (NTHREADS * 4))  // 32 float4 loads per thread

__device__ __forceinline__ float wave_max(float v) {
    // wave32 butterfly reduce
    #pragma unroll
    for (int off = 16; off > 0; off >>= 1)
        v = fmaxf(v, __shfl_xor(v, off, 32));
    return v;
}

__device__ __forceinline__ float wave_sum(float v) {
    #pragma unroll
    for (int off = 16; off > 0; off >>= 1)
        v += __shfl_xor(v, off, 32);
    return v;
}

__global__ void __launch_bounds__(NTHREADS)
row_lse_kernel(const float* __restrict__ start_logits,
               const float* __restrict__ end_logits,
               const int*   __restrict__ start_pos,
               const int*   __restrict__ end_pos,
               float*       __restrict__ partials) {
    __shared__ float red[NWAVES];
    __shared__ float bcast;

    const int b     = blockIdx.x;        // 0..511
    const int which = b & 1;             // 0 = start, 1 = end
    const int row   = b >> 1;

    const float* __restrict__ x =
        (which ? end_logits : start_logits) + (size_t)row * SEQ;
    const int pos = which ? end_pos[row] : start_pos[row];

    const int t    = threadIdx.x;
    const int lane = t & 31;
    const int wid  = t >> 5;

    const float4* __restrict__ x4 = reinterpret_cast<const float4*>(x);

    // ---- pass 1: row max (streams 128 KB from HBM, coalesced b128 loads) ----
    float m = -3.402823466e38f;
    #pragma unroll 8
    for (int c = 0; c < CHUNKS; ++c) {
        const float4 v = x4[c * NTHREADS + t];
        m = fmaxf(m, fmaxf(fmaxf(v.x, v.y), fmaxf(v.z, v.w)));
    }
    m = wave_max(m);
    if (lane == 0) red[wid] = m;
    __syncthreads();
    if (wid == 0) {
        float r = (lane < NWAVES) ? red[lane] : -3.402823466e38f;
        r = wave_max(r);
        if (lane == 0) bcast = r;
    }
    __syncthreads();
    const float rowmax = bcast;
    __syncthreads();

    // ---- pass 2: sum exp(x - max); re-read hits L2 (row is 128 KB) ----
    float s = 0.0f;
    #pragma unroll 8
    for (int c = 0; c < CHUNKS; ++c) {
        const float4 v = x4[c * NTHREADS + t];
        s += __expf(v.x - rowmax) + __expf(v.y - rowmax) +
             __expf(v.z - rowmax) + __expf(v.w - rowmax);
    }
    s = wave_sum(s);
    if (lane == 0) red[wid] = s;
    __syncthreads();
    if (wid == 0) {
        float r = (lane < NWAVES) ? red[lane] : 0.0f;
        r = wave_sum(r);
        if (lane == 0) {
            // partial contribution: lse - x[pos]
            partials[b] = rowmax + __logf(r) - x[pos];
        }
    }
}

// Reduce 512 f32 partials to the scalar loss with 8 accumulating
// V_WMMA_F32_16X16X4_F32 ops (B = all-ones => D rows accumulate row-sums).
__global__ void __launch_bounds__(32)
final_reduce_wmma(const float* __restrict__ partials, float* __restrict__ out) {
    const int lane = threadIdx.x;    // one full wave32, EXEC all ones

    v2f ones;
    ones.x = 1.0f; ones.y = 1.0f;

    v8f c = {};
    #pragma unroll
    for (int i = 0; i < 8; ++i) {
        // 32-bit A-matrix 16x4 layout: lane L, vgpr j -> A[L%16][2*(L>=16)+j]
        const int base = i * 64 + (lane & 15) * 4 + ((lane >> 4) << 1);
        v2f a;
        a.x = partials[base];
        a.y = partials[base + 1];
        // D = A * ones + C : every column n of D holds the 16 row-sums
        c = __builtin_amdgcn_wmma_f32_16x16x4_f32(
                /*neg_a=*/false, a, /*neg_b=*/false, ones,
                /*c_mod=*/(short)0, c, /*reuse_a=*/false, /*reuse_b=*/false);
    }

    // column 0 lives in lanes 0 (M=0..7 across c[0..7]) and 16 (M=8..15)
    float s = c[0] + c[1] + c[2] + c[3] + c[4] + c[5] + c[6] + c[7];
    s += __shfl_xor(s, 16, 32);
    if (lane == 0) {
        *out = s * (1.0f / 512.0f);   // 1 / (2*B), B = 256
    }
}

extern "C" void kernel_launch(void* const* d_in, const int* in_sizes, int n_in,
                              void* d_out, int out_size, void* d_ws, size_t ws_size,
                              hipStream_t stream) {
    const float* start_logits = (const float*)d_in[0];  // [256, 32768, 1]
    const float* end_logits   = (const float*)d_in[1];  // [256, 32768, 1]
    const int*   start_pos    = (const int*)d_in[2];    // [256]
    const int*   end_pos      = (const int*)d_in[3];    // [256]
    float*       out          = (float*)d_out;          // scalar
    float*       partials     = (float*)d_ws;           // 512 floats

    row_lse_kernel<<<512, NTHREADS, 0, stream>>>(start_logits, end_logits,
                                                 start_pos, end_pos, partials);
    final_reduce_wmma<<<1, 32, 0, stream>>>(partials, out);
}